// MyModel_26938034880747
// MI455X (gfx1250) — compile-verified
//
#include <hip/hip_runtime.h>
#include <hip/hip_bf16.h>
#include <math.h>

// Problem dimensions (fixed by the reference).
#define T_STEPS 20
#define N_NODES 20000
#define N_EDGES 320000
#define IN_CH   128
#define HID_CH  256
#define OUT_CH  256
#define NCLS    16

typedef __attribute__((ext_vector_type(2))) float v2f;
typedef __attribute__((ext_vector_type(8))) float v8f;

// ---------------------------------------------------------------------------
// Zero-fill
// ---------------------------------------------------------------------------
__global__ void fill_kernel(float* __restrict__ p, float v, int n) {
    int i = blockIdx.x * blockDim.x + threadIdx.x;
    if (i < n) p[i] = v;
}

// ---------------------------------------------------------------------------
// Edge scatter: agg[dst] += x[src] (float4 per thread), cnt[dst] += 1
// ---------------------------------------------------------------------------
template<int C>
__global__ void scatter_kernel(const float* __restrict__ x,
                               const int*  __restrict__ src,
                               const int*  __restrict__ dst,
                               float* __restrict__ agg,
                               float* __restrict__ cnt) {
    const int per_edge = C / 4;
    int idx = blockIdx.x * blockDim.x + threadIdx.x;
    if (idx >= N_EDGES * per_edge) return;
    int e  = idx / per_edge;
    int c4 = idx - e * per_edge;
    int s = src[e];
    int d = dst[e];
    float4 v = *(const float4*)(x + (long long)s * C + c4 * 4);
    float* a = agg + (long long)d * C + c4 * 4;
    atomicAdd(a + 0, v.x);
    atomicAdd(a + 1, v.y);
    atomicAdd(a + 2, v.z);
    atomicAdd(a + 3, v.w);
    if (c4 == 0) atomicAdd(cnt + d, 1.0f);
}

// ---------------------------------------------------------------------------
// Fused SAGE layer: out = [relu]( (agg/max(cnt,1)) @ Wl^T + xin @ Wr^T + b )
// One wave32 per 16x64 output macro-tile: 4 column tiles share one A fragment
// (4x WMMA per A load -> 1.25 loads/WMMA, 4x less A traffic).
//   A 16x4 layout : lane(L&15)=M, (L>>4)*2 = K pair base, 2 VGPRs
//   B 4x16 layout : lane(L&15)=N, same K pair grouping, 2 VGPRs
//   C/D 16x16     : vgpr i -> M = i + 8*(L>>4), N = L&15
// Grid sized exactly (N_NODES % 16 == 0) so EXEC is all-ones at every WMMA.
// ---------------------------------------------------------------------------
template<int K, int J, bool RELU>
__global__ void sage_gemm_kernel(const float* __restrict__ agg,
                                 const float* __restrict__ cnt,
                                 const float* __restrict__ xin,
                                 const float* __restrict__ Wl,   // [J,K]
                                 const float* __restrict__ Wr,   // [J,K]
                                 const float* __restrict__ bias, // [J]
                                 float* __restrict__ out) {      // [N,J]
    constexpr int JG = 4;  // column tiles per wave
    const int lane = threadIdx.x & 31;
    const int wave = threadIdx.x >> 5;
    const int tile = blockIdx.x * 4 + wave;        // 4 waves / block
    const int colGroups = J / (16 * JG);
    const int rowBase = (tile / colGroups) * 16;
    const int colBase = (tile % colGroups) * (16 * JG);

    const int m     = lane & 15;        // A row / B,C column within tile
    const int khalf = (lane >> 4) * 2;  // this lane-group's K offset

    const int arow = rowBase + m;       // node row this lane loads for A
    const float invc = __builtin_amdgcn_rcpf(fmaxf(cnt[arow], 1.0f));

    const float* aptr = agg + (long long)arow * K + khalf;
    const float* xptr = xin + (long long)arow * K + khalf;
    const float* wl[JG];
    const float* wr[JG];
#pragma unroll
    for (int g = 0; g < JG; ++g) {
        const int wrow = colBase + g * 16 + m;     // weight row (out channel)
        wl[g] = Wl + (long long)wrow * K + khalf;
        wr[g] = Wr + (long long)wrow * K + khalf;
        __builtin_prefetch(wl[g], 0, 3);           // global_prefetch_b8
        __builtin_prefetch(wr[g], 0, 3);
    }

    v8f acc[JG] = {};
    // mean @ Wl^T  (mean = agg * rcp(cnt) applied on the A fragment)
    for (int k = 0; k < K; k += 4) {
        float2 av = *(const float2*)(aptr + k);
        v2f a;
        a.x = av.x * invc;
        a.y = av.y * invc;
#pragma unroll
        for (int g = 0; g < JG; ++g) {
            float2 bv = *(const float2*)(wl[g] + k);
            v2f b; b.x = bv.x; b.y = bv.y;
            acc[g] = __builtin_amdgcn_wmma_f32_16x16x4_f32(
                false, a, false, b, (short)0, acc[g], false, false);
        }
    }
    // xin @ Wr^T
    for (int k = 0; k < K; k += 4) {
        float2 av = *(const float2*)(xptr + k);
        v2f a; a.x = av.x; a.y = av.y;
#pragma unroll
        for (int g = 0; g < JG; ++g) {
            float2 bv = *(const float2*)(wr[g] + k);
            v2f b; b.x = bv.x; b.y = bv.y;
            acc[g] = __builtin_amdgcn_wmma_f32_16x16x4_f32(
                false, a, false, b, (short)0, acc[g], false, false);
        }
    }

    const int rhalf = (lane >> 4) * 8;
#pragma unroll
    for (int g = 0; g < JG; ++g) {
        const float bv = bias[colBase + g * 16 + m];
#pragma unroll
        for (int i = 0; i < 8; ++i) {
            float v = acc[g][i] + bv;
            if (RELU) v = fmaxf(v, 0.0f);
            out[(long long)(rowBase + rhalf + i) * J + colBase + g * 16 + m] = v;
        }
    }
}

// ---------------------------------------------------------------------------
// Global max-pool over nodes: seqOut[j] = max_n h[n][j]
// ---------------------------------------------------------------------------
__global__ void maxpool_kernel(const float* __restrict__ h,
                               float* __restrict__ seqOut) {
    __shared__ float red[256];
    const int j = blockIdx.x;  // channel
    float m = -INFINITY;
    for (int n = threadIdx.x; n < N_NODES; n += blockDim.x)
        m = fmaxf(m, h[(long long)n * OUT_CH + j]);
    red[threadIdx.x] = m;
    __syncthreads();
    for (int s = 128; s > 0; s >>= 1) {
        if (threadIdx.x < s)
            red[threadIdx.x] = fmaxf(red[threadIdx.x], red[threadIdx.x + s]);
        __syncthreads();
    }
    if (threadIdx.x == 0) seqOut[j] = red[0];
}

// ---------------------------------------------------------------------------
// GRU over T steps + log_softmax. Tiny: one block.
// ---------------------------------------------------------------------------
__global__ void gru_kernel(const float* __restrict__ seq,  // [T, OUT_CH]
                           const float* __restrict__ Wih,  // [48, OUT_CH]
                           const float* __restrict__ Whh,  // [48, 16]
                           const float* __restrict__ bih,  // [48]
                           const float* __restrict__ bhh,  // [48]
                           float* __restrict__ out) {      // [T, 16]
    __shared__ float h[NCLS];
    __shared__ float gi[3 * NCLS];
    __shared__ float gh[3 * NCLS];
    __shared__ float hist[T_STEPS * NCLS];
    const int t = threadIdx.x;
    if (t < NCLS) h[t] = 0.0f;
    __syncthreads();

    for (int step = 0; step < T_STEPS; ++step) {
        const float* xt = seq + step * OUT_CH;
        if (t < 3 * NCLS) {
            float si = bih[t];
            for (int k = 0; k < OUT_CH; ++k) si += Wih[t * OUT_CH + k] * xt[k];
            gi[t] = si;
            float sh = bhh[t];
            for (int k = 0; k < NCLS; ++k) sh += Whh[t * NCLS + k] * h[k];
            gh[t] = sh;
        }
        __syncthreads();
        if (t < NCLS) {
            float r = 1.0f / (1.0f + expf(-(gi[t] + gh[t])));
            float z = 1.0f / (1.0f + expf(-(gi[NCLS + t] + gh[NCLS + t])));
            float n = tanhf(gi[2 * NCLS + t] + r * gh[2 * NCLS + t]);
            float hn = (1.0f - z) * n + z * h[t];
            h[t] = hn;
            hist[step * NCLS + t] = hn;
        }
        __syncthreads();
    }

    if (t < T_STEPS) {
        float m = -INFINITY;
        for (int j = 0; j < NCLS; ++j) m = fmaxf(m, hist[t * NCLS + j]);
        float s = 0.0f;
        for (int j = 0; j < NCLS; ++j) s += expf(hist[t * NCLS + j] - m);
        float lse = m + logf(s);
        for (int j = 0; j < NCLS; ++j)
            out[t * NCLS + j] = hist[t * NCLS + j] - lse;
    }
}

// ---------------------------------------------------------------------------
// kernel_launch
// ---------------------------------------------------------------------------
extern "C" void kernel_launch(void* const* d_in, const int* in_sizes, int n_in,
                              void* d_out, int out_size, void* d_ws, size_t ws_size,
                              hipStream_t stream) {
    const float* x   = (const float*)d_in[0];   // [T, N, 128]
    const int*   ei  = (const int*)  d_in[1];   // [T, 2, E]
    // d_in[2] = batch (all zeros, single graph) -- unused
    const float* W1l = (const float*)d_in[3];   // [256, 128]
    const float* b1  = (const float*)d_in[4];   // [256]
    const float* W1r = (const float*)d_in[5];   // [256, 128]
    const float* W2l = (const float*)d_in[6];   // [256, 256]
    const float* b2  = (const float*)d_in[7];   // [256]
    const float* W2r = (const float*)d_in[8];   // [256, 256]
    const float* Wih = (const float*)d_in[9];   // [48, 256]
    const float* Whh = (const float*)d_in[10];  // [48, 16]
    const float* bih = (const float*)d_in[11];  // [48]
    const float* bhh = (const float*)d_in[12];  // [48]
    float* out = (float*)d_out;                 // [T, 16]

    char* ws = (char*)d_ws;
    float* agg = (float*)ws; ws += sizeof(float) * (size_t)N_NODES * HID_CH;
    float* cnt = (float*)ws; ws += sizeof(float) * (size_t)N_NODES;
    float* h1  = (float*)ws; ws += sizeof(float) * (size_t)N_NODES * HID_CH;
    float* h2  = (float*)ws; ws += sizeof(float) * (size_t)N_NODES * OUT_CH;
    float* seq = (float*)ws; ws += sizeof(float) * (size_t)T_STEPS * OUT_CH;

    // waves = rowTiles * colGroups; 4 waves (128 threads) per block
    const int gemm1_blocks = (N_NODES / 16) * (HID_CH / 64) / 4;
    const int gemm2_blocks = (N_NODES / 16) * (OUT_CH / 64) / 4;
    const int sc1_blocks = (N_EDGES * (IN_CH  / 4) + 255) / 256;
    const int sc2_blocks = (N_EDGES * (HID_CH / 4) + 255) / 256;

    for (int t = 0; t < T_STEPS; ++t) {
        const float* xt  = x  + (size_t)t * N_NODES * IN_CH;
        const int*   src = ei + (size_t)t * 2 * N_EDGES;
        const int*   dst = src + N_EDGES;

        // ---- SAGE layer 1 (128 -> 256, relu) ----
        fill_kernel<<<(N_NODES * IN_CH + 255) / 256, 256, 0, stream>>>(
            agg, 0.0f, N_NODES * IN_CH);
        fill_kernel<<<(N_NODES + 255) / 256, 256, 0, stream>>>(
            cnt, 0.0f, N_NODES);
        scatter_kernel<IN_CH><<<sc1_blocks, 256, 0, stream>>>(
            xt, src, dst, agg, cnt);
        sage_gemm_kernel<IN_CH, HID_CH, true><<<gemm1_blocks, 128, 0, stream>>>(
            agg, cnt, xt, W1l, W1r, b1, h1);

        // ---- SAGE layer 2 (256 -> 256, no relu) ----
        fill_kernel<<<(N_NODES * HID_CH + 255) / 256, 256, 0, stream>>>(
            agg, 0.0f, N_NODES * HID_CH);
        fill_kernel<<<(N_NODES + 255) / 256, 256, 0, stream>>>(
            cnt, 0.0f, N_NODES);
        scatter_kernel<HID_CH><<<sc2_blocks, 256, 0, stream>>>(
            h1, src, dst, agg, cnt);
        sage_gemm_kernel<HID_CH, OUT_CH, false><<<gemm2_blocks, 128, 0, stream>>>(
            agg, cnt, h1, W2l, W2r, b2, h2);

        // ---- global max pool ----
        maxpool_kernel<<<OUT_CH, 256, 0, stream>>>(h2, seq + t * OUT_CH);
    }

    // ---- GRU head + log_softmax ----
    gru_kernel<<<1, 64, 0, stream>>>(seq, Wih, Whh, bih, bhh, out);
}